// Simplex4Net_9242769621293
// MI455X (gfx1250) — compile-verified
//
#include <hip/hip_runtime.h>
#include <hip/hip_bf16.h>

#define NB  32   // batch
#define NP  64   // points
#define ND  64   // hidden
#define SCALE 0.125f  // 1/sqrt(64)

typedef __attribute__((ext_vector_type(2))) float v2f;
typedef __attribute__((ext_vector_type(8))) float v8f;

// workspace layout (float offsets)
constexpr int Q_OFF   = 0;          // (32,64,64) queries
constexpr int KJ_OFF  = 131072;     // keys_j
constexpr int KK_OFF  = 262144;     // keys_k
constexpr int KL_OFF  = 393216;     // keys_l
constexpr int AJK_OFF = 524288;     // (32,64,64) att_jk (pre-scaled)
constexpr int AJL_OFF = 655360;
constexpr int AKL_OFF = 786432;
constexpr int FEAT_OFF = 917504;    // (32,64) anchor feats [b][i]

// ---------------- Kernel A: projections (B blocks x 64 threads) ----------------
__global__ void proj_kernel(const float* __restrict__ pc,
                            const float* __restrict__ Wq,  const float* __restrict__ bq,
                            const float* __restrict__ Wk1, const float* __restrict__ bk1,
                            const float* __restrict__ Wk2, const float* __restrict__ bk2,
                            const float* __restrict__ Wk3, const float* __restrict__ bk3,
                            float* __restrict__ ws) {
    const int b = blockIdx.x;
    const int n = threadIdx.x;           // 0..63
    const float px = pc[(b * NP + n) * 3 + 0];
    const float py = pc[(b * NP + n) * 3 + 1];
    const float pz = pc[(b * NP + n) * 3 + 2];
    float* q  = ws + Q_OFF  + (b * NP + n) * ND;
    float* kj = ws + KJ_OFF + (b * NP + n) * ND;
    float* kk = ws + KK_OFF + (b * NP + n) * ND;
    float* kl = ws + KL_OFF + (b * NP + n) * ND;
    for (int d = 0; d < ND; ++d) {
        q[d]  = fmaf(px, Wq [d], fmaf(py, Wq [ND + d], fmaf(pz, Wq [2 * ND + d], bq [d])));
        kj[d] = fmaf(px, Wk1[d], fmaf(py, Wk1[ND + d], fmaf(pz, Wk1[2 * ND + d], bk1[d])));
        kk[d] = fmaf(px, Wk2[d], fmaf(py, Wk2[ND + d], fmaf(pz, Wk2[2 * ND + d], bk2[d])));
        kl[d] = fmaf(px, Wk3[d], fmaf(py, Wk3[ND + d], fmaf(pz, Wk3[2 * ND + d], bk3[d])));
    }
}

// ------- Kernel B: key-key attention GEMMs via V_WMMA_F32_16X16X4_F32 ----------
// att_jk = keys_j @ keys_k^T * scale (and _jl, _kl). Per batch: 3 matrices x 16
// (16x16) tiles = 48 tile jobs over 8 waves. Exact f32 matmul, K=64 in steps of 4.
__global__ __launch_bounds__(256) void att_kernel(float* __restrict__ ws) {
    const int b    = blockIdx.x;
    const int lane = threadIdx.x & 31;
    const int wave = threadIdx.x >> 5;   // 0..7
    const int h    = lane >> 4;          // lane-half: selects K pair {2h,2h+1}
    const int m16  = lane & 15;

    const float* kjb = ws + KJ_OFF + b * NP * ND;
    const float* kkb = ws + KK_OFF + b * NP * ND;
    const float* klb = ws + KL_OFF + b * NP * ND;

    for (int job = wave; job < 48; job += 8) {      // uniform per wave (EXEC all-1)
        const int m  = job >> 4;                    // which matrix
        const int t  = job & 15;
        const int jt = t >> 2, kt = t & 3;
        const float* Aa = (m == 2) ? kkb : kjb;
        const float* Bb = (m == 0) ? kkb : klb;
        float* Dp = ws + ((m == 0) ? AJK_OFF : (m == 1) ? AJL_OFF : AKL_OFF) + b * NP * NP;

        const float* arow = Aa + (jt * 16 + m16) * ND;   // A: M=m16 within tile
        const float* brow = Bb + (kt * 16 + m16) * ND;   // B = keys^T: N=m16
        v8f c = {0.f, 0.f, 0.f, 0.f, 0.f, 0.f, 0.f, 0.f};
        #pragma unroll
        for (int d0 = 0; d0 < ND; d0 += 4) {
            v2f av = *(const v2f*)(arow + d0 + 2 * h);   // K = d0+2h, d0+2h+1
            v2f bv = *(const v2f*)(brow + d0 + 2 * h);
            c = __builtin_amdgcn_wmma_f32_16x16x4_f32(
                    false, av, false, bv, (short)0, c, false, false);
        }
        // C/D layout: VGPR r holds M = r + 8*h, N = m16
        #pragma unroll
        for (int r = 0; r < 8; ++r) {
            const int jrow = jt * 16 + r + 8 * h;
            Dp[jrow * NP + kt * 16 + m16] = c[r] * SCALE;
        }
    }
}

// ---------- Kernel C: per-(b,i) energy/gate/det reduction (2048 blocks) ---------
// LDS tiles are fused at staging time:
//   s_jk[j][k] = scale*att_jk + att_ij[j]
//   s_kl[k][l] = scale*att_kl + att_il[l]
// so the inner j-loop does only 3 LDS reads: disp[j] (b128 broadcast),
// s_jk[j][k] (b32 broadcast, k wave-uniform), s_jl[j][l] (b32 stride-1).
__global__ __launch_bounds__(256) void energy_kernel(const float* __restrict__ pc,
                                                     float* __restrict__ ws) {
    __shared__ float  s_jk[NP * NP];
    __shared__ float  s_jl[NP * NP];
    __shared__ float  s_kl[NP * NP];
    __shared__ float4 s_disp[NP];
    __shared__ float  s_aij[NP], s_aik[NP], s_ail[NP];
    __shared__ float  s_red[256];

    const int b   = blockIdx.x >> 6;
    const int i   = blockIdx.x & 63;
    const int tid = threadIdx.x;

    // phase 1: disp vectors + q_i . keys dot products
    const float pix = pc[(b * NP + i) * 3 + 0];
    const float piy = pc[(b * NP + i) * 3 + 1];
    const float piz = pc[(b * NP + i) * 3 + 2];
    const float* qi = ws + Q_OFF + (b * NP + i) * ND;

    if (tid < 192) {
        const int t   = tid & 63;
        const int grp = tid >> 6;   // 0: att_ij, 1: att_ik, 2: att_il
        const float* kr = ws + ((grp == 0) ? KJ_OFF : (grp == 1) ? KK_OFF : KL_OFF)
                          + (b * NP + t) * ND;
        float s = 0.f;
        for (int d = 0; d < ND; ++d) s = fmaf(qi[d], kr[d], s);
        s *= SCALE;
        if (grp == 0) {
            s_aij[t] = s;
            s_disp[t] = make_float4(pc[(b * NP + t) * 3 + 0] - pix,
                                    pc[(b * NP + t) * 3 + 1] - piy,
                                    pc[(b * NP + t) * 3 + 2] - piz, 0.f);
        } else if (grp == 1) s_aik[t] = s;
        else                 s_ail[t] = s;
    }
    __syncthreads();

    // phase 2: stage att matrices into LDS with aij/ail folded in (float4)
    const float4* gjk = (const float4*)(ws + AJK_OFF + b * NP * NP);
    const float4* gjl = (const float4*)(ws + AJL_OFF + b * NP * NP);
    const float4* gkl = (const float4*)(ws + AKL_OFF + b * NP * NP);
    for (int idx = tid; idx < (NP * NP) / 4; idx += 256) {
        const int row = idx >> 4;          // j (for jk/jl) or k (for kl)
        const int c0  = (idx & 15) * 4;    // first column of this float4
        float4 vjk = gjk[idx];
        const float aijr = s_aij[row];
        vjk.x += aijr; vjk.y += aijr; vjk.z += aijr; vjk.w += aijr;
        ((float4*)s_jk)[idx] = vjk;
        ((float4*)s_jl)[idx] = gjl[idx];
        float4 vkl = gkl[idx];
        vkl.x += s_ail[c0 + 0]; vkl.y += s_ail[c0 + 1];
        vkl.z += s_ail[c0 + 2]; vkl.w += s_ail[c0 + 3];
        ((float4*)s_kl)[idx] = vkl;
    }
    __syncthreads();

    // phase 3: main (k,l) x j loop
    float acc = 0.f;
    for (int idx = tid; idx < NP * NP; idx += 256) {
        const int k = idx >> 6;          // uniform within a wave -> LDS broadcast
        const int l = idx & 63;          // lane-contiguous -> conflict-free
        const float4 dk = s_disp[k];
        const float4 dl = s_disp[l];
        const float cx = fmaf(dk.y, dl.z, -dk.z * dl.y);
        const float cy = fmaf(dk.z, dl.x, -dk.x * dl.z);
        const float cz = fmaf(dk.x, dl.y, -dk.y * dl.x);
        const float base = s_aik[k] + s_kl[k * NP + l];   // ail[l] already folded
        #pragma unroll 8
        for (int j = 0; j < NP; ++j) {
            const float4 dj  = s_disp[j];                           // ds b128 broadcast
            const float  det = fmaf(dj.x, cx, fmaf(dj.y, cy, dj.z * cz));
            const float  e   = base + s_jk[j * NP + k] + s_jl[j * NP + l];
            const float  ex  = __expf(-e);
            const float  g   = __builtin_amdgcn_rcpf(1.0f + ex);    // sigmoid
            acc = fmaf(det * det, g, acc);
        }
    }

    s_red[tid] = acc;
    __syncthreads();
    for (int s = 128; s > 0; s >>= 1) {
        if (tid < s) s_red[tid] += s_red[tid + s];
        __syncthreads();
    }
    if (tid == 0)
        ws[FEAT_OFF + b * NP + i] = s_red[0] * (1.0f / 262144.0f);  // /N^3
}

// ---------------- Kernel D: pooling + gelu MLP head (1 block x 32) --------------
__global__ void head_kernel(const float* __restrict__ ws,
                            const float* __restrict__ W1, const float* __restrict__ b1,
                            const float* __restrict__ W2, const float* __restrict__ b2,
                            float* __restrict__ out) {
    const int b = threadIdx.x;           // 0..31
    const float* f = ws + FEAT_OFF + b * NP;
    float s = 0.f;
    for (int i = 0; i < NP; ++i) s += f[i];
    const float pooled = s * (1.0f / NP);
    float o = 0.f;
    for (int m = 0; m < 32; ++m) {
        const float z = fmaf(pooled, W1[m], b1[m]);
        // tanh-approx gelu (flax default)
        const float t = tanhf(0.7978845608028654f * (z + 0.044715f * z * z * z));
        const float g = 0.5f * z * (1.0f + t);
        o = fmaf(g, W2[m], o);
    }
    out[b] = o + b2[0];
}

extern "C" void kernel_launch(void* const* d_in, const int* in_sizes, int n_in,
                              void* d_out, int out_size, void* d_ws, size_t ws_size,
                              hipStream_t stream) {
    (void)in_sizes; (void)n_in; (void)out_size; (void)ws_size;
    const float* pc  = (const float*)d_in[0];
    const float* Wq  = (const float*)d_in[1];
    const float* bq  = (const float*)d_in[2];
    const float* Wk1 = (const float*)d_in[3];
    const float* bk1 = (const float*)d_in[4];
    const float* Wk2 = (const float*)d_in[5];
    const float* bk2 = (const float*)d_in[6];
    const float* Wk3 = (const float*)d_in[7];
    const float* bk3 = (const float*)d_in[8];
    const float* W1  = (const float*)d_in[9];
    const float* b1  = (const float*)d_in[10];
    const float* W2  = (const float*)d_in[11];
    const float* b2  = (const float*)d_in[12];
    float* ws  = (float*)d_ws;
    float* out = (float*)d_out;

    proj_kernel<<<NB, NP, 0, stream>>>(pc, Wq, bq, Wk1, bk1, Wk2, bk2, Wk3, bk3, ws);
    att_kernel<<<NB, 256, 0, stream>>>(ws);
    energy_kernel<<<NB * NP, 256, 0, stream>>>(pc, ws);
    head_kernel<<<1, 32, 0, stream>>>(ws, W1, b1, W2, b2, out);
}